// DKVMN_58944131170323
// MI455X (gfx1250) — compile-verified
//
#include <hip/hip_runtime.h>
#include <hip/hip_bf16.h>
#include <math.h>

typedef __attribute__((ext_vector_type(2))) float v2f;
typedef __attribute__((ext_vector_type(8))) float v8f;

#define EMB 64
#define NUM_C 200
#define NSZ 4096
#define BB 64
#define LL 200
#define SIZE_M 64
#define N_NODES 201
#define BL (BB*LL)

// ---------------- CDNA5 async global->LDS helpers ----------------
__device__ __forceinline__ void async_copy_b128(void* lds_dst, const void* gsrc) {
  unsigned loff = (unsigned)(uintptr_t)(__attribute__((address_space(3))) void*)lds_dst;
  asm volatile("global_load_async_to_lds_b128 %0, %1, off"
               :: "v"(loff), "v"(gsrc) : "memory");
}
__device__ __forceinline__ void wait_async_all() {
  asm volatile("s_wait_asynccnt 0x0" ::: "memory");
}

// ---------------- activation helper ----------------
__device__ __forceinline__ float activate(float x, int act) {
  switch (act) {
    case 1: return x > 0.f ? x : 0.f;             // relu
    case 2: return 1.f / (1.f + __expf(-x));      // sigmoid
    case 3: return tanhf(x);                      // tanh
    default: return x;
  }
}

// ---------------- generic fp32 WMMA GEMM ----------------
// C[M,N] = act(A[M,K] @ W[K,N] + bias[N] + add[M,N]),  K in {8..128}, mult of 4.
// 256 threads = 8 waves; W panel (K x 64 cols) staged in LDS, each wave owns a
// 16x64 strip of C. Edge handling by index clamping (no divergent loads):
// out-of-range A rows / W cols only feed C entries we never store.
__global__ __launch_bounds__(256) void gemm_wmma_f32(
    const float* __restrict__ A, const float* __restrict__ W,
    const float* __restrict__ bias, const float* __restrict__ add,
    float* __restrict__ C, int M, int N, int K, int act)
{
  __shared__ float lW[128 * 64];         // up to 32 KB (K<=128)
  const int tid  = threadIdx.x;
  const int lane = tid & 31;
  const int wave = tid >> 5;             // 0..7
  const int half = lane >> 4;            // K sub-pair select
  const int l15  = lane & 15;
  const int m0   = blockIdx.x * 128 + wave * 16;
  const int n0   = blockIdx.y * 64;
  const int ncols = (N - n0) < 64 ? (N - n0) : 64;

  // stage W panel (cols clamped into valid range -> unconditional loads)
  for (int i = tid; i < K * 64; i += 256) {
    const int kk = i >> 6, c = i & 63;
    const int cc = n0 + (c < ncols ? c : ncols - 1);
    lW[i] = W[(size_t)kk * N + cc];
  }
  __syncthreads();

  const int arow = (m0 + l15) < M ? (m0 + l15) : (M - 1);   // clamped
  const float* aPtr = A + (size_t)arow * K;

  v8f acc[4] = {{}, {}, {}, {}};
  for (int k = 0; k < K; k += 4) {
    const v2f a = *(const v2f*)(aPtr + k + half * 2);       // aligned b64
#pragma unroll
    for (int nt = 0; nt < 4; ++nt) {
      v2f b;
      const int col = nt * 16 + l15;
      b.x = lW[(k + half * 2    ) * 64 + col];
      b.y = lW[(k + half * 2 + 1) * 64 + col];
      acc[nt] = __builtin_amdgcn_wmma_f32_16x16x4_f32(
          false, a, false, b, (short)0, acc[nt], false, false);
    }
  }

#pragma unroll
  for (int nt = 0; nt < 4; ++nt) {
    const int col = n0 + nt * 16 + l15;
    if (col >= N) continue;
    const float bv = bias ? bias[col] : 0.f;
#pragma unroll
    for (int r = 0; r < 8; ++r) {
      const int row = m0 + r + half * 8;
      if (row >= M) continue;
      float v = acc[nt][r] + bv;
      if (add) v += add[(size_t)row * N + col];
      C[(size_t)row * N + col] = activate(v, act);
    }
  }
}

// ---------------- big GEMM vs dense G (4096x4096) ----------------
// C[4096,64] = opt_relu(G[4096,4096] @ X[4096,64]); LDS-staged tiles;
// X panel staged via CDNA5 async global->LDS (ASYNCcnt) path.
#define GKC 64
__global__ __launch_bounds__(256) void gemm_G_wmma(
    const float* __restrict__ G, const float* __restrict__ X,
    float* __restrict__ C, int relu)
{
  __shared__ float lA[128 * GKC];   // 32 KB: G rows m0..m0+127, cols kc..kc+63
  __shared__ float lB[GKC * 64];    // 16 KB: X rows kc..kc+63
  const int tid  = threadIdx.x;
  const int lane = tid & 31;
  const int wave = tid >> 5;        // 0..7, each owns 16 rows x 64 cols
  const int m0   = blockIdx.x * 128;
  const int half = lane >> 4;
  const int l15  = lane & 15;

  v8f acc[4] = {{}, {}, {}, {}};

  for (int kc = 0; kc < NSZ; kc += GKC) {
    __syncthreads();
    // stage X chunk (contiguous 64x64 block) via async global->LDS b128
    {
      const float4* src = (const float4*)(X + (size_t)kc * 64);
      for (int i = tid; i < (GKC * 64) / 4; i += 256)
        async_copy_b128(&((float4*)lB)[i], &src[i]);
    }
    // stage G chunk (128 rows x 64 cols) through VGPRs (b128)
    for (int i = tid; i < 128 * 16; i += 256) {
      const int r = i >> 4, c4 = i & 15;
      ((float4*)lA)[r * 16 + c4] =
          *(const float4*)(G + (size_t)(m0 + r) * NSZ + kc + c4 * 4);
    }
    // prefetch next G chunk into L2 while we compute
    if (kc + GKC < NSZ && tid < 128)
      __builtin_prefetch(G + (size_t)(m0 + tid) * NSZ + kc + GKC, 0, 1);
    wait_async_all();
    __syncthreads();

    const float* aRow = &lA[(wave * 16 + l15) * GKC];
    for (int kk = 0; kk < GKC; kk += 4) {
      v2f a;
      a.x = aRow[kk + half * 2];
      a.y = aRow[kk + half * 2 + 1];
#pragma unroll
      for (int nt = 0; nt < 4; ++nt) {
        v2f b;
        const int col = nt * 16 + l15;
        b.x = lB[(kk + half * 2    ) * 64 + col];
        b.y = lB[(kk + half * 2 + 1) * 64 + col];
        acc[nt] = __builtin_amdgcn_wmma_f32_16x16x4_f32(
            false, a, false, b, (short)0, acc[nt], false, false);
      }
    }
  }

#pragma unroll
  for (int nt = 0; nt < 4; ++nt) {
    const int col = nt * 16 + l15;
#pragma unroll
    for (int r = 0; r < 8; ++r) {
      const int row = m0 + wave * 16 + r + half * 8;
      float v = acc[nt][r];
      if (relu) v = v > 0.f ? v : 0.f;
      C[(size_t)row * 64 + col] = v;
    }
  }
}

// ---------------- 64x64 transpose (Mk -> Mk^T) ----------------
__global__ void transpose64(const float* __restrict__ Min, float* __restrict__ Mout)
{
  int i = blockIdx.x * 256 + threadIdx.x;
  if (i < 64 * 64) { int r = i >> 6, c = i & 63; Mout[c * 64 + r] = Min[i]; }
}

// ---------------- gated fusion (gather + sigmoid gate) ----------------
// mode 0: ctx = SE[student-1]   (h-side); mode 1: ctx = pooled[b] (d-side)
__global__ void gated_kernel(
    const float* __restrict__ SE, const float* __restrict__ pooled,
    const int* __restrict__ student, const int* __restrict__ skill,
    const int* __restrict__ answer,
    const float* __restrict__ k_emb, const float* __restrict__ v_emb,
    const float* __restrict__ gkW, const float* __restrict__ gkb,
    const float* __restrict__ gvW, const float* __restrict__ gvb,
    float* __restrict__ Kout, float* __restrict__ Vout, int mode)
{
  int p = blockIdx.x * blockDim.x + threadIdx.x;
  if (p >= BL) return;
  const int b = p / LL;
  const float* ctx = mode ? (pooled + (size_t)b * EMB)
                          : (SE + (size_t)(student[p] - 1) * EMB);
  const int sk = skill[p];
  const int an = answer[p];
  const int ax = (an == 2) ? 1 : an;
  const float* ke = k_emb + (size_t)sk * EMB;
  const float* ve = v_emb + (size_t)(sk + NUM_C * ax) * EMB;
  float gk = gkb[0], gv = gvb[0];
  for (int e = 0; e < EMB; ++e) { gk += ctx[e] * gkW[e];      gv += ctx[e] * gvW[e]; }
  for (int e = 0; e < EMB; ++e) { gk += ke[e] * gkW[EMB + e]; gv += ve[e] * gvW[EMB + e]; }
  gk = 1.f / (1.f + __expf(-gk));
  gv = 1.f / (1.f + __expf(-gv));
  for (int e = 0; e < EMB; ++e) {
    Kout[(size_t)p * EMB + e] = gk * ctx[e] + (1.f - gk) * ke[e];
    Vout[(size_t)p * EMB + e] = gv * ctx[e] + (1.f - gv) * ve[e];
  }
}

// ---------------- row softmax over 64 ----------------
__global__ void softmax64(float* __restrict__ Wl, int n)
{
  int p = blockIdx.x * blockDim.x + threadIdx.x;
  if (p >= n) return;
  float* r = Wl + (size_t)p * 64;
  float mx = r[0];
  for (int i = 1; i < 64; ++i) mx = fmaxf(mx, r[i]);
  float s = 0.f;
  for (int i = 0; i < 64; ++i) { float v = __expf(r[i] - mx); r[i] = v; s += v; }
  const float inv = 1.f / s;
  for (int i = 0; i < 64; ++i) r[i] *= inv;
}

// ---------------- DKVMN sequential scan (Mv in LDS) ----------------
__global__ __launch_bounds__(256) void dkvmn_scan(
    const float* __restrict__ Mv0, const float* __restrict__ Wt,
    const float* __restrict__ Et, const float* __restrict__ At,
    float* __restrict__ reads)
{
  __shared__ float Mv[SIZE_M * EMB];
  __shared__ float wv[SIZE_M], ev[EMB], av[EMB];
  __shared__ float psum[4][EMB];
  const int b = blockIdx.x;
  const int tid = threadIdx.x;
  const int e = tid & 63, g = tid >> 6;   // 4 groups of 64
  for (int i = tid; i < SIZE_M * EMB; i += 256) Mv[i] = Mv0[i];
  for (int t = 0; t < LL; ++t) {
    __syncthreads();
    const size_t base = ((size_t)b * LL + t) * EMB;
    if (tid < 64) { wv[tid] = Wt[base + tid]; ev[tid] = Et[base + tid]; av[tid] = At[base + tid]; }
    __syncthreads();
    float s = 0.f;
    for (int m = g * 16; m < g * 16 + 16; ++m) s += wv[m] * Mv[m * EMB + e];
    psum[g][e] = s;
    __syncthreads();
    if (g == 0) reads[base + e] = psum[0][e] + psum[1][e] + psum[2][e] + psum[3][e];
    const float ee = ev[e], aa = av[e];
    for (int m = g * 16; m < g * 16 + 16; ++m) {
      const float wm = wv[m];
      Mv[m * EMB + e] = Mv[m * EMB + e] * (1.f - wm * ee) + wm * aa;
    }
  }
}

// ---------------- GCN (two layers, per-batch workgroup) ----------------
__global__ __launch_bounds__(256) void gcn_kernel(
    const float* __restrict__ xw1,   // [201,8] = k_emb @ g1_W
    const float* __restrict__ g1b, const float* __restrict__ g2W,
    const int* __restrict__ skill, float* __restrict__ all_out /*B,201,64*/)
{
  __shared__ float deg[N_NODES], dinv[N_NODES];
  __shared__ float h1[N_NODES * 8];
  __shared__ float g2w[8 * EMB];
  __shared__ int se[LL - 1], de[LL - 1];
  const int b = blockIdx.x, tid = threadIdx.x;
  const int* sb = skill + (size_t)b * LL;
  float* outb = all_out + (size_t)b * N_NODES * EMB;
  for (int i = tid; i < N_NODES; i += 256) deg[i] = 1.f;           // self loop
  for (int i = tid; i < N_NODES * 8; i += 256) h1[i] = 0.f;
  for (int i = tid; i < 8 * EMB; i += 256) g2w[i] = g2W[i];
  for (int i = tid; i < LL - 1; i += 256) { se[i] = sb[i]; de[i] = sb[i + 1]; }
  for (int i = tid; i < N_NODES * EMB; i += 256) outb[i] = 0.f;
  __syncthreads();
  for (int i = tid; i < LL - 1; i += 256) atomicAdd(&deg[de[i]], 1.f);
  __syncthreads();
  for (int i = tid; i < N_NODES; i += 256) dinv[i] = deg[i] > 0.f ? rsqrtf(deg[i]) : 0.f;
  __syncthreads();
  // layer 1: self loops + edges
  for (int i = tid; i < N_NODES * 8; i += 256) {
    int n = i >> 3, f = i & 7;
    atomicAdd(&h1[i], dinv[n] * dinv[n] * xw1[n * 8 + f]);
  }
  __syncthreads();
  for (int idx = tid; idx < (LL - 1) * 8; idx += 256) {
    int ei = idx >> 3, f = idx & 7;
    int s = se[ei], d = de[ei];
    atomicAdd(&h1[d * 8 + f], dinv[s] * dinv[d] * xw1[s * 8 + f]);
  }
  __syncthreads();
  for (int i = tid; i < N_NODES * 8; i += 256) {
    int f = i & 7;
    float v = h1[i] + g1b[f];
    h1[i] = v > 0.f ? v : 0.f;
  }
  __syncthreads();
  // layer 2: y = h1 @ g2W computed on the fly, aggregated with global atomics
  for (int i = tid; i < N_NODES * EMB; i += 256) {
    int n = i >> 6, e2 = i & 63;
    float y = 0.f;
    for (int f = 0; f < 8; ++f) y += h1[n * 8 + f] * g2w[f * EMB + e2];
    atomicAdd(&outb[i], dinv[n] * dinv[n] * y);
  }
  for (int idx = tid; idx < (LL - 1) * EMB; idx += 256) {
    int ei = idx >> 6, e2 = idx & 63;
    int s = se[ei], d = de[ei];
    float y = 0.f;
    for (int f = 0; f < 8; ++f) y += h1[s * 8 + f] * g2w[f * EMB + e2];
    atomicAdd(&outb[d * EMB + e2], dinv[s] * dinv[d] * y);
  }
}

// ---------------- masked pooled mean (incl. g2 bias) ----------------
__global__ void pool_kernel(const float* __restrict__ all_out,
                            const float* __restrict__ g2b,
                            const int* __restrict__ skill,
                            const int* __restrict__ answer,
                            float* __restrict__ pooled)
{
  const int b = blockIdx.x, e = threadIdx.x;   // 64 threads
  float s = 0.f;
  for (int l = 0; l < LL; ++l) {
    if (answer[b * LL + l] != 2) {
      const int n = skill[b * LL + l];
      s += all_out[((size_t)b * N_NODES + n) * EMB + e] + g2b[e];
    }
  }
  pooled[b * EMB + e] = s / (float)LL;
}

// ---------------- concat [A|Bm] -> C (n rows of 64+64) ----------------
__global__ void concat_kernel(const float* __restrict__ A, const float* __restrict__ Bm,
                              float* __restrict__ C, int nElem)
{
  int i = blockIdx.x * blockDim.x + threadIdx.x;
  if (i < nElem) {
    int p = i >> 6, e = i & 63;
    C[(size_t)p * 128 + e]      = A[i];
    C[(size_t)p * 128 + 64 + e] = Bm[i];
  }
}

// ---------------- theta mix -> concat [theta*h | (1-theta)*dft] ----------------
__global__ void mix_kernel(const float* __restrict__ theta, const float* __restrict__ h,
                           const float* __restrict__ dft, float* __restrict__ cc, int nElem)
{
  int i = blockIdx.x * blockDim.x + threadIdx.x;
  if (i < nElem) {
    int p = i >> 6, e = i & 63;
    const float t = theta[i];
    cc[(size_t)p * 128 + e]      = t * h[i];
    cc[(size_t)p * 128 + 64 + e] = (1.f - t) * dft[i];
  }
}

// ---------------- next-skill prediction gather ----------------
__global__ void pred_kernel(const float* __restrict__ lh, const float* __restrict__ ld,
                            const float* __restrict__ le, const int* __restrict__ skill,
                            float* __restrict__ out)
{
  int i = blockIdx.x * blockDim.x + threadIdx.x;   // over B*(L-1)
  const int NP = BB * (LL - 1);
  if (i >= NP) return;
  const int b = i / (LL - 1), l = i % (LL - 1);
  const int ns = skill[b * LL + l + 1];
  const int idx = ns < NUM_C - 1 ? ns : NUM_C - 1;
  const size_t off = ((size_t)b * LL + l) * NUM_C + idx;
  const bool ok = ns < NUM_C;
  out[i]          = ok ? lh[off] : 0.f;
  out[NP + i]     = ok ? ld[off] : 0.f;
  out[2 * NP + i] = ok ? le[off] : 0.f;
}

// ---------------- host orchestration ----------------
static inline void launch_gemm(const float* A, const float* W, const float* bias,
                               const float* add, float* C, int M, int N, int K,
                               int act, hipStream_t s)
{
  dim3 grid((M + 127) / 128, (N + 63) / 64);
  hipLaunchKernelGGL(gemm_wmma_f32, grid, dim3(256), 0, s, A, W, bias, add, C, M, N, K, act);
}

extern "C" void kernel_launch(void* const* d_in, const int* in_sizes, int n_in,
                              void* d_out, int out_size, void* d_ws, size_t ws_size,
                              hipStream_t stream)
{
  auto f = [&](int i) { return (const float*)d_in[i]; };
  // params in setup_inputs dict order
  const float *hg1_W = f(0),  *hg1_b = f(1),  *hg2_W = f(2),  *hg2_b = f(3);
  const float *gkhW  = f(4),  *gkhb  = f(5),  *gvhW  = f(6),  *gvhb  = f(7);
  const float *gkdW  = f(8),  *gkdb  = f(9),  *gvdW  = f(10), *gvdb  = f(11);
  const float *eh_W  = f(12), *eh_b  = f(13), *ah_W  = f(14), *ah_b  = f(15);
  const float *ed_W  = f(16), *ed_b  = f(17), *ad_W  = f(18), *ad_b  = f(19);
  const float *fh_W  = f(20), *fh_b  = f(21), *fd_W  = f(22), *fd_b  = f(23);
  const float *h_W   = f(24), *h_b   = f(25), *d_W   = f(26), *d_b   = f(27);
  const float *w1_W  = f(28), *w1_b  = f(29), *w2_W  = f(30), *w2_b  = f(31);
  const float *ensW  = f(32), *ensB  = f(33), *g1_W  = f(34), *g1_b  = f(35);
  const float *g2_W  = f(36), *g2_b  = f(37);
  const float *stu   = f(38), *k_emb = f(39), *v_emb = f(40);
  const float *Mk    = f(41), *Mv0   = f(42), *G     = f(43);
  const int *student = (const int*)d_in[44];
  const int *skill   = (const int*)d_in[45];
  const int *answer  = (const int*)d_in[46];
  float* out = (float*)d_out;

  // carve workspace (floats)
  float* w = (float*)d_ws;
  size_t off = 0;
  auto alloc = [&](size_t n) { float* p = w + off; off += n; return p; };
  float* T1   = alloc((size_t)NSZ * 64);
  float* Xb   = alloc((size_t)NSZ * 64);
  float* T2   = alloc((size_t)NSZ * 64);
  float* SE   = alloc((size_t)NSZ * 64);
  float* MkT  = alloc(64 * 64);
  float* xw1  = alloc(256 * 8);
  float* allo = alloc((size_t)BB * N_NODES * EMB);
  float* pool = alloc((size_t)BB * EMB);
  float* Kh   = alloc((size_t)BL * EMB);
  float* Vh   = alloc((size_t)BL * EMB);
  float* Wh   = alloc((size_t)BL * EMB);
  float* Eh   = alloc((size_t)BL * EMB);
  float* Ah   = alloc((size_t)BL * EMB);
  float* Rh   = alloc((size_t)BL * EMB);
  float* CCh  = alloc((size_t)BL * 128);
  float* Hh   = alloc((size_t)BL * EMB);
  float* LGh  = alloc((size_t)BL * NUM_C);
  float* Kd   = alloc((size_t)BL * EMB);
  float* Vd   = alloc((size_t)BL * EMB);
  float* Wd   = alloc((size_t)BL * EMB);
  float* Ed   = alloc((size_t)BL * EMB);
  float* Ad   = alloc((size_t)BL * EMB);
  float* Rd   = alloc((size_t)BL * EMB);
  float* CCd  = alloc((size_t)BL * 128);
  float* Dft  = alloc((size_t)BL * EMB);
  float* LGd  = alloc((size_t)BL * NUM_C);
  float* Uw   = alloc((size_t)BL * EMB);
  float* Th   = alloc((size_t)BL * EMB);
  float* H2D2 = alloc((size_t)BL * 128);
  float* LGe  = alloc((size_t)BL * NUM_C);
  (void)ws_size; (void)in_sizes; (void)n_in; (void)out_size;

  const int thr = 256;
  const int posBlocks = (BL + thr - 1) / thr;
  const int elemBlocks = (BL * EMB + thr - 1) / thr;

  // Mk^T for w = softmax(k @ Mk^T)
  hipLaunchKernelGGL(transpose64, dim3(16), dim3(256), 0, stream, Mk, MkT);

  // student-graph embedding: SE = G @ ((relu(G @ (stu@hg1+b)))@hg2+b)
  launch_gemm(stu, hg1_W, hg1_b, nullptr, T1, NSZ, 64, 64, 0, stream);
  hipLaunchKernelGGL(gemm_G_wmma, dim3(NSZ / 128), dim3(256), 0, stream, G, T1, Xb, 1);
  launch_gemm(Xb, hg2_W, hg2_b, nullptr, T2, NSZ, 64, 64, 0, stream);
  hipLaunchKernelGGL(gemm_G_wmma, dim3(NSZ / 128), dim3(256), 0, stream, G, T2, SE, 0);

  // GCN branch: xw1 = k_emb @ g1_W, then 2-layer GCN per batch, then pooled mean
  launch_gemm(k_emb, g1_W, nullptr, nullptr, xw1, N_NODES, 8, 64, 0, stream);
  hipLaunchKernelGGL(gcn_kernel, dim3(BB), dim3(256), 0, stream, xw1, g1_b, g2_W, skill, allo);
  hipLaunchKernelGGL(pool_kernel, dim3(BB), dim3(64), 0, stream, allo, g2_b, skill, answer, pool);

  // gated key/value (h-side uses SE gather, d-side uses pooled ctx)
  hipLaunchKernelGGL(gated_kernel, dim3(posBlocks), dim3(thr), 0, stream,
                     SE, pool, student, skill, answer, k_emb, v_emb,
                     gkhW, gkhb, gvhW, gvhb, Kh, Vh, 0);
  hipLaunchKernelGGL(gated_kernel, dim3(posBlocks), dim3(thr), 0, stream,
                     SE, pool, student, skill, answer, k_emb, v_emb,
                     gkdW, gkdb, gvdW, gvdb, Kd, Vd, 1);

  // attention weights + erase/add vectors
  launch_gemm(Kh, MkT, nullptr, nullptr, Wh, BL, 64, 64, 0, stream);
  hipLaunchKernelGGL(softmax64, dim3(posBlocks), dim3(thr), 0, stream, Wh, BL);
  launch_gemm(Kd, MkT, nullptr, nullptr, Wd, BL, 64, 64, 0, stream);
  hipLaunchKernelGGL(softmax64, dim3(posBlocks), dim3(thr), 0, stream, Wd, BL);
  launch_gemm(Vh, eh_W, eh_b, nullptr, Eh, BL, 64, 64, 2, stream);
  launch_gemm(Vh, ah_W, ah_b, nullptr, Ah, BL, 64, 64, 3, stream);
  launch_gemm(Vd, ed_W, ed_b, nullptr, Ed, BL, 64, 64, 2, stream);
  launch_gemm(Vd, ad_W, ad_b, nullptr, Ad, BL, 64, 64, 3, stream);

  // sequential DKVMN memory scans
  hipLaunchKernelGGL(dkvmn_scan, dim3(BB), dim3(256), 0, stream, Mv0, Wh, Eh, Ah, Rh);
  hipLaunchKernelGGL(dkvmn_scan, dim3(BB), dim3(256), 0, stream, Mv0, Wd, Ed, Ad, Rd);

  // feature heads
  hipLaunchKernelGGL(concat_kernel, dim3(elemBlocks), dim3(thr), 0, stream, Rh, Kh, CCh, BL * EMB);
  launch_gemm(CCh, fh_W, fh_b, nullptr, Hh, BL, 64, 128, 3, stream);
  hipLaunchKernelGGL(concat_kernel, dim3(elemBlocks), dim3(thr), 0, stream, Rd, Kd, CCd, BL * EMB);
  launch_gemm(CCd, fd_W, fd_b, nullptr, Dft, BL, 64, 128, 3, stream);
  launch_gemm(Hh, h_W, h_b, nullptr, LGh, BL, NUM_C, 64, 0, stream);
  launch_gemm(Dft, d_W, d_b, nullptr, LGd, BL, NUM_C, 64, 0, stream);

  // ensemble: theta = sigmoid(h@w1 + b1 + dft@w2 + b2)
  launch_gemm(Dft, w2_W, w2_b, nullptr, Uw, BL, 64, 64, 0, stream);
  launch_gemm(Hh, w1_W, w1_b, Uw, Th, BL, 64, 64, 2, stream);
  hipLaunchKernelGGL(mix_kernel, dim3(elemBlocks), dim3(thr), 0, stream, Th, Hh, Dft, H2D2, BL * EMB);
  launch_gemm(H2D2, ensW, ensB, nullptr, LGe, BL, NUM_C, 128, 0, stream);

  // final next-skill gather -> out (3 concatenated (B,L-1) arrays)
  const int NP = BB * (LL - 1);
  hipLaunchKernelGGL(pred_kernel, dim3((NP + thr - 1) / thr), dim3(thr), 0, stream,
                     LGh, LGd, LGe, skill, out);
}